// Dense2DSpatialTransformer_16269336117379
// MI455X (gfx1250) — compile-verified
//
#include <hip/hip_runtime.h>

typedef float v4f __attribute__((ext_vector_type(4)));
typedef int   v4i __attribute__((vector_size(16)));   // builtin's data type

constexpr int Bn   = 16;
constexpr int Hn   = 1024;
constexpr int Wn   = 1024;
constexpr int TPB  = 256;   // 8 wave32 waves
constexpr int ROWS = 8;     // rows of 1024 px per block, pipelined
constexpr int PPT  = 4;     // pixels per thread -> b128 traffic

// ---- CDNA5 async global->LDS copy (ASYNCcnt path) ---------------------------
__device__ __forceinline__ void async_load_b128(const float* gsrc, float* ldst) {
#if __has_builtin(__builtin_amdgcn_global_load_async_to_lds_b128)
    __builtin_amdgcn_global_load_async_to_lds_b128(
        (__attribute__((address_space(1))) v4i*)(gsrc),   // global src (AS1)
        (__attribute__((address_space(3))) v4i*)(ldst),   // LDS dst (AS3)
        /*imm offset*/0, /*cpol*/0);
#else
    unsigned loff = (unsigned)(unsigned long long)
        (__attribute__((address_space(3))) float*)ldst;
    asm volatile("global_load_async_to_lds_b128 %0, %1, off"
                 :: "v"(loff), "v"(gsrc) : "memory");
#endif
}

template <int N>
__device__ __forceinline__ void wait_async() {
#if __has_builtin(__builtin_amdgcn_s_wait_asynccnt)
    __builtin_amdgcn_s_wait_asynccnt(N);
#else
    asm volatile("s_wait_asynccnt %0" :: "i"(N) : "memory");
#endif
}
// -----------------------------------------------------------------------------

extern "C" __global__ void __launch_bounds__(TPB)
warp_bilinear_kernel(const float* __restrict__ img,   // (B,1,H,W)
                     const float* __restrict__ flow,  // (B,2,H,W)
                     float* __restrict__ out)         // (B,1,H,W)
{
    __shared__ float sH[2][Wn];   // dH row, double-buffered
    __shared__ float sW[2][Wn];   // dW row, double-buffered

    const int t    = threadIdx.x;
    const int row0 = blockIdx.x * ROWS;        // global row id in [0, B*H)
    const int b    = row0 / Hn;                // H % ROWS == 0 -> no batch straddle
    const int h0   = row0 % Hn;
    const int c0   = t * PPT;                  // this thread's 4 columns

    const size_t plane = (size_t)Hn * Wn;
    const float* imgB  = img  + (size_t)b * plane;
    const float* dHrow = flow + (size_t)b * 2 * plane + (size_t)h0 * Wn;
    const float* dWrow = dHrow + plane;        // second channel of this batch
    float*       outR  = out  + (size_t)b * plane + (size_t)h0 * Wn;

    // prologue: DMA stage 0 flow row into LDS
    async_load_b128(dHrow + c0, &sH[0][c0]);
    async_load_b128(dWrow + c0, &sW[0][c0]);

    for (int s = 0; s < ROWS; ++s) {
        const int buf = s & 1;

        if (s + 1 < ROWS) {
            const int nb = (s + 1) & 1;
            async_load_b128(dHrow + (size_t)(s + 1) * Wn + c0, &sH[nb][c0]);
            async_load_b128(dWrow + (size_t)(s + 1) * Wn + c0, &sW[nb][c0]);
            // likely gather target for next stage: same row neighborhood
            const int hn = h0 + s + 1;
            if (hn < Hn) __builtin_prefetch(imgB + (size_t)hn * Wn + c0, 0, 3);
            wait_async<2>();   // stage s done (in-order), stage s+1 in flight
        } else {
            wait_async<0>();
        }

        const int h = h0 + s;
        const v4f dh4 = *(const v4f*)&sH[buf][c0];
        const v4f dw4 = *(const v4f*)&sW[buf][c0];

        v4f res;
#pragma unroll
        for (int j = 0; j < PPT; ++j) {
            // padded coordinates (reference pads by 1 with edge mode)
            const float Hu = dh4[j] + (float)h + 1.0f;
            const float Wu = dw4[j] + (float)(c0 + j) + 1.0f;
            const int hf = (int)floorf(Hu);
            const int wf = (int)floorf(Wu);
            // clamp in padded coords [0, H+1] / [0, W+1]
            const int hf_c = min(max(hf,     0), Hn + 1);
            const int hc_c = min(max(hf + 1, 0), Hn + 1);
            const int wf_c = min(max(wf,     0), Wn + 1);
            const int wc_c = min(max(wf + 1, 0), Wn + 1);
            // weights use the CLAMPED ceil coords (matches reference exactly)
            const float dh = (float)hc_c - Hu;
            const float dw = (float)wc_c - Wu;
            // padded -> source coords (edge padding == clamp(y-1, 0, H-1))
            const int y0 = min(max(hf_c - 1, 0), Hn - 1);
            const int y1 = min(max(hc_c - 1, 0), Hn - 1);
            const int x0 = min(max(wf_c - 1, 0), Wn - 1);
            const int x1 = min(max(wc_c - 1, 0), Wn - 1);

            const float* r0 = imgB + (size_t)y0 * Wn;
            const float* r1 = imgB + (size_t)y1 * Wn;
            const float v00 = r0[x0];
            const float v10 = r0[x1];
            const float v01 = r1[x0];
            const float v11 = r1[x1];

            const float w00 = dh * dw;
            const float w10 = dh * (1.0f - dw);
            const float w01 = (1.0f - dh) * dw;
            const float w11 = (1.0f - dw) * (1.0f - dh);
            res[j] = fmaf(v00, w00, fmaf(v10, w10, fmaf(v01, w01, v11 * w11)));
        }
        // write-once output: non-temporal b128 store
        __builtin_nontemporal_store(res, (v4f*)(outR + (size_t)s * Wn + c0));
    }
}

extern "C" void kernel_launch(void* const* d_in, const int* in_sizes, int n_in,
                              void* d_out, int out_size, void* d_ws, size_t ws_size,
                              hipStream_t stream) {
    const float* img  = (const float*)d_in[0];   // (16,1,1024,1024) f32
    const float* flow = (const float*)d_in[1];   // (16,2,1024,1024) f32
    float* out        = (float*)d_out;           // (16,1,1024,1024) f32

    const int nblocks = (Bn * Hn) / ROWS;        // 2048
    warp_bilinear_kernel<<<nblocks, TPB, 0, stream>>>(img, flow, out);
}